// CounterfactualReasoning_10952166605058
// MI455X (gfx1250) — compile-verified
//
#include <hip/hip_runtime.h>

// CDNA5 / gfx1250, wave32. One 1024-thread block (32 waves) owns 16 rows of
// sp/m/v/s fully register-resident (as float2 pairs) across all 50 Adam steps.
// Per-step z[i] = sp[i,:]·W via v_wmma_f32_16x16x4_f32: wave w owns columns
// [32w,32w+32); the v2f state pairs ARE the ISA A-matrix 16x4 f32 operand
// layout (V0=K0|K2, V1=K1|K3, M=lane%16), and W stored identically is the
// B-matrix 4x16 layout with all 16 N-columns equal. 8 chained WMMAs/wave/step.
// Elementwise Adam runs on the packed-f32 pipe (v_pk_*), with v_rcp/v_sqrt
// instead of IEEE divides (workload is VALU-bound: ~10 GFLOP vs 67 MB DRAM).

typedef __attribute__((ext_vector_type(2))) float v2f;
typedef __attribute__((ext_vector_type(4))) float v4f;
typedef __attribute__((ext_vector_type(8))) float v8f;

#define BROWS   8192
#define DCOLS   1024
#define NSTEPS  50
#define PSTRIDE 40   // floats per row in LDS partial buffer (16B-aligned, padded)

__launch_bounds__(1024, 1)
__global__ void adam_fused_wmma(const float* __restrict__ s,
                                const float* __restrict__ tconf,
                                const float* __restrict__ W,
                                const float* __restrict__ bias,
                                float* __restrict__ out) {
    __shared__ float lds_part[16 * PSTRIDE];  // [row][wave] z partials (padded)
    __shared__ float lds_gz[16];              // per-row dL/dz broadcast
    __shared__ float lds_tc[16];              // per-row target confidence

    const int tid  = threadIdx.x;
    const int wv   = tid >> 5;        // wave 0..31 -> column slice [32*wv, +32)
    const int lane = tid & 31;
    const int hi   = lane >> 4;       // half-wave (K-pair select)
    const int r    = lane & 15;       // row within the 16-row tile
    const int rowg = blockIdx.x * 16 + r;

    if (tid < 16) lds_tc[tid] = tconf[blockIdx.x * 16 + tid];
    const float bval = bias[0];

    // State as v2f pairs. Pair j holds columns {c, c+1}, c = 32*wv + 4*j + 2*hi.
    // This is exactly the WMMA f32 A/B operand layout for K-chunk j.
    v2f sp2[8], mm2[8], vv2[8], ss2[8], Wv2[8];
    const int colbase = wv * 32 + 2 * hi;
    const long rbase = (long)rowg * DCOLS;
    #pragma unroll
    for (int j = 0; j < 8; ++j) {
        const int c = colbase + 4 * j;
        v2f sv = *(const v2f*)(s + rbase + c);
        v2f wl = *(const v2f*)(W + c);
        ss2[j] = sv;  sp2[j] = sv;  Wv2[j] = wl;
        mm2[j] = (v2f)(0.0f);  vv2[j] = (v2f)(0.0f);
    }
    __syncthreads();

    const float B1 = 0.9f, B2 = 0.999f, LR = 0.01f, EPS = 1e-8f;
    const float OMB1 = 1.0f - B1, OMB2 = 1.0f - B2;
    const float COEF  = 2.0f / 8192.0f;                 // 2/n_mse
    const float INVL1 = 1.0f / (8192.0f * 1024.0f);     // 1/n_l1
    float p1 = 1.0f, p2 = 1.0f;                         // B1^t, B2^t running

    for (int step = 0; step < NSTEPS; ++step) {
        // Bias-correction scalars (independent of gz: off the critical path).
        p1 *= B1; p2 *= B2;
        const float stepa = LR * __builtin_amdgcn_rcpf(1.0f - p1);  // lr/(1-b1^t)
        const float c2    = __builtin_amdgcn_rcpf(1.0f - p2);       // 1/(1-b2^t)

        // ---- z partial for this wave's 32 columns via 8 chained WMMAs ----
        v8f acc = {0.f, 0.f, 0.f, 0.f, 0.f, 0.f, 0.f, 0.f};
        #pragma unroll
        for (int j = 0; j < 8; ++j) {
            acc = __builtin_amdgcn_wmma_f32_16x16x4_f32(
                      false, sp2[j], false, Wv2[j], (short)0, acc, false, false);
        }
        // C layout: VGPR p -> row p (lanes 0-15) / row p+8 (lanes 16-31),
        // all N-columns identical. One lane per half-wave exports partials
        // into the [row][wave] transposed buffer.
        if (r == 0) {
            #pragma unroll
            for (int p = 0; p < 8; ++p)
                lds_part[(hi * 8 + p) * PSTRIDE + wv] = acc[p];
        }
        __syncthreads();

        // ---- deterministic cross-wave reduce + dL/dz (one thread per row) ----
        if (tid < 16) {
            float z = bval;
            const float* rowp = &lds_part[tid * PSTRIDE];
            #pragma unroll
            for (int j = 0; j < 8; ++j) {
                v4f x = *(const v4f*)(rowp + 4 * j);   // ds_load_b128
                z += (x.x + x.y) + (x.z + x.w);
            }
            float pr = 1.0f / (1.0f + expf(-z));                  // sigmoid
            lds_gz[tid] = COEF * (pr - lds_tc[tid]) * pr * (1.0f - pr);
        }
        __syncthreads();
        const float gz = lds_gz[r];

        // ---- elementwise Adam update: packed-f32 pipe + rcp/sqrt trans ----
        #pragma unroll
        for (int j = 0; j < 8; ++j) {
            v2f d = sp2[j] - ss2[j];
            v2f sg;
            sg.x = (float)(d.x > 0.0f) - (float)(d.x < 0.0f);
            sg.y = (float)(d.y > 0.0f) - (float)(d.y < 0.0f);
            v2f g  = gz * Wv2[j] + sg * INVL1;          // v_pk_mul + v_pk_fma
            mm2[j] = B1 * mm2[j] + OMB1 * g;            // v_pk_*
            vv2[j] = B2 * vv2[j] + (OMB2 * g) * g;      // v_pk_*
            v2f den;
            den.x = __builtin_amdgcn_rcpf(__builtin_amdgcn_sqrtf(vv2[j].x * c2) + EPS);
            den.y = __builtin_amdgcn_rcpf(__builtin_amdgcn_sqrtf(vv2[j].y * c2) + EPS);
            sp2[j] = sp2[j] - (stepa * mm2[j]) * den;   // v_pk_*
        }
    }

    // ---- write final sp ----
    #pragma unroll
    for (int j = 0; j < 8; ++j)
        *(v2f*)(out + rbase + colbase + 4 * j) = sp2[j];
}

extern "C" void kernel_launch(void* const* d_in, const int* in_sizes, int n_in,
                              void* d_out, int out_size, void* d_ws, size_t ws_size,
                              hipStream_t stream) {
    const float* s  = (const float*)d_in[0];   // [8192,1024]
    const float* tc = (const float*)d_in[1];   // [8192,1]
    const float* W  = (const float*)d_in[2];   // [1,1024]
    const float* b  = (const float*)d_in[3];   // [1]
    float* out = (float*)d_out;                // [8192,1024]
    (void)in_sizes; (void)n_in; (void)d_ws; (void)ws_size; (void)out_size;

    adam_fused_wmma<<<BROWS / 16, 1024, 0, stream>>>(s, tc, W, b, out);
}